// LCN_9182640079272
// MI455X (gfx1250) — compile-verified
//
#include <hip/hip_runtime.h>

// LCN (locally-connected) + decoder for MI455X (gfx1250), wave32.
// Stage 1: per-patch GEMM  M=512(batch) x N=16(filters) x K=784 using
//          V_WMMA_F32_16X16X4_F32 (exact fp32), weights staged in LDS.
// Stage 2: (512 x 1600) @ (1600 x 10)^T + bias, wave-reduced.

typedef float v2f __attribute__((ext_vector_type(2)));
typedef float v8f __attribute__((ext_vector_type(8)));

namespace lcn {
constexpr int kB   = 512;
constexpr int kH   = 280;
constexpr int kW   = 280;
constexpr int kKS  = 28;
constexpr int kHS  = 10;
constexpr int kWS  = 10;
constexpr int kP   = 100;   // patches
constexpr int kF   = 16;    // filters per patch
constexpr int kK   = 784;   // KS*KS reduction length
constexpr int kOUT = 10;
constexpr int kWavesPerBlock = 8;   // 256 threads
constexpr int kFP  = kF * kP;       // 1600
}

__global__ __launch_bounds__(256)
void lcn_stage1_wmma(const float* __restrict__ x,
                     const float* __restrict__ weight,
                     const float* __restrict__ bias,
                     float* __restrict__ ybuf)
{
    using namespace lcn;
    // 16 filters x 784 floats for this patch = 50176 B of LDS.
    __shared__ float sw[kF * kK];

    const int p   = blockIdx.x;          // patch id 0..99
    const int ph  = p / kWS;
    const int pw  = p % kWS;
    const int tid = threadIdx.x;

    // Cooperative float4 staging of weight[(f*P + p)*K + k] into LDS.
    constexpr int kVec4 = (kF * kK) / 4; // 3136
    for (int g = tid; g < kVec4; g += 256) {
        const int f = g / (kK / 4);
        const int r = g % (kK / 4);
        const float4* src =
            reinterpret_cast<const float4*>(weight + (size_t)(f * kP + p) * kK) + r;
        reinterpret_cast<float4*>(sw + f * kK)[r] = *src;
    }
    __syncthreads();

    const int wave = tid >> 5;
    const int lane = tid & 31;
    const int half = lane >> 4;   // 0 -> K pair {0,1}; 1 -> K pair {2,3}
    const int mn   = lane & 15;   // A: batch row; B: filter column

    const int b0 = (blockIdx.y * kWavesPerBlock + wave) * 16;

    // Per-lane base of this (batch,row / patch) in x: x[b][ph*28+kh][pw*28+kw]
    const float* xb = x + (size_t)(b0 + mn) * (kH * kW)
                        + (size_t)(ph * kKS) * kW + pw * kKS;
    const float* wb = sw + mn * kK;

    v8f acc0 = {};
    v8f acc1 = {};

    for (int kh = 0; kh < kKS; ++kh) {
        const float* xrow = xb + kh * kW;
        const float* wrow = wb + kh * kKS;
        #pragma unroll
        for (int kwb = 0; kwb < kKS; kwb += 4) {
            // K-group [k0, k0+3] always lies inside one 28-float row (28 % 4 == 0),
            // so each lane does one aligned float2 load from x and one from LDS.
            v2f a = *reinterpret_cast<const v2f*>(xrow + kwb + half * 2);
            v2f b = *reinterpret_cast<const v2f*>(wrow + kwb + half * 2);
            if ((kwb >> 2) & 1)
                acc1 = __builtin_amdgcn_wmma_f32_16x16x4_f32(
                    false, a, false, b, (short)0, acc1, false, false);
            else
                acc0 = __builtin_amdgcn_wmma_f32_16x16x4_f32(
                    false, a, false, b, (short)0, acc0, false, false);
        }
    }

    // C/D layout: VGPR j, lanes 0-15 -> M=j, lanes 16-31 -> M=j+8; N = lane&15.
    const float bv = bias[mn * kP + p];
    #pragma unroll
    for (int j = 0; j < 8; ++j) {
        float y = acc0[j] + acc1[j] + bv;
        y = y > 0.0f ? y : 0.0f;
        const int b = b0 + j + half * 8;
        ybuf[(size_t)b * kFP + mn * kP + p] = y;  // (B, F*P) with flat f*100+p
    }
}

__global__ __launch_bounds__(320)
void lcn_stage2_decode(const float* __restrict__ ybuf,
                       const float* __restrict__ dec_w,
                       const float* __restrict__ dec_b,
                       float* __restrict__ out)
{
    using namespace lcn;
    const int b = blockIdx.x;
    const int o = threadIdx.x >> 5;   // output 0..9, one wave each
    const int l = threadIdx.x & 31;

    const float* yr = ybuf  + (size_t)b * kFP;
    const float* wr = dec_w + (size_t)o * kFP;

    float acc = 0.0f;
    #pragma unroll 5
    for (int i = l; i < kFP; i += 32)
        acc += yr[i] * wr[i];

    #pragma unroll
    for (int m = 16; m >= 1; m >>= 1)
        acc += __shfl_xor(acc, m, 32);

    if (l == 0)
        out[b * kOUT + o] = acc + dec_b[o];
}

extern "C" void kernel_launch(void* const* d_in, const int* in_sizes, int n_in,
                              void* d_out, int out_size, void* d_ws, size_t ws_size,
                              hipStream_t stream)
{
    using namespace lcn;
    const float* x      = (const float*)d_in[0];
    const float* weight = (const float*)d_in[1];
    const float* bias   = (const float*)d_in[2];
    const float* dec_w  = (const float*)d_in[3];
    const float* dec_b  = (const float*)d_in[4];
    float* out  = (float*)d_out;
    float* ybuf = (float*)d_ws;   // 512*1600 floats = 3.28 MB scratch

    dim3 g1(kP, kB / (16 * kWavesPerBlock));   // (100, 4)
    lcn_stage1_wmma<<<g1, 256, 0, stream>>>(x, weight, bias, ybuf);
    lcn_stage2_decode<<<kB, 320, 0, stream>>>(ybuf, dec_w, dec_b, out);
}